// SurgicalTriXLayer_5162550690212
// MI455X (gfx1250) — compile-verified
//
#include <hip/hip_runtime.h>
#include <hip/hip_bf16.h>

#define D_MODEL   1024
#define N_TILES   16
#define N_CLASSES 1024
#define BATCH     65536
#define THRESH    0.3f
#define MAXT      (BATCH / 16 + N_TILES)   // 4112 max row-tiles after padding

typedef float v2f __attribute__((ext_vector_type(2)));
typedef float v8f __attribute__((ext_vector_type(8)));

// ---- workspace layout (int32 offsets into d_ws) ----
#define WS_COUNTS   0                       // [16]
#define WS_CURSORS  16                      // [16]
#define WS_NUMTILES 32                      // [1]
#define WS_EXPERTOF 64                      // [BATCH]
#define WS_PERM     (64 + BATCH)            // [BATCH + 256] (padded buckets)
#define WS_TILEEXP  (WS_PERM + BATCH + 256) // [MAXT]
#define WS_TILEBASE (WS_TILEEXP + MAXT)     // [MAXT]

__global__ void init_kernel(int* __restrict__ ws) {
    if (threadIdx.x < N_TILES) ws[WS_COUNTS + threadIdx.x] = 0;
}

// ---------------------------------------------------------------------------
// Routing: scores = x @ ternary(sigs)^T via fp32 WMMA, argmax, atomic counts.
// One block = 128 tokens (8 waves x 16-token tiles).
// ---------------------------------------------------------------------------
__global__ __launch_bounds__(256) void route_kernel(
    const float* __restrict__ x, const float* __restrict__ sigs_raw,
    float* __restrict__ idx_out, int* __restrict__ ws)
{
    // XOR-swizzled ternary signatures: row stride 1024 floats would put all
    // 16 lanes on LDS bank 0; swizzling the column with (expert*4) spreads
    // the b64 fragment reads across all 64 banks. Reused for score transpose.
    __shared__ float sq[N_TILES * D_MODEL];   // 64 KB
    const int tid = threadIdx.x;

    for (int i = tid; i < N_TILES * D_MODEL; i += 256) {
        int t = i >> 10, k = i & (D_MODEL - 1);
        float s = sigs_raw[i];
        float q = (s > THRESH) ? 1.0f : ((s < -THRESH) ? -1.0f : 0.0f);
        sq[(t << 10) + (k ^ (t << 2))] = q;
    }
    __syncthreads();

    const int wave = tid >> 5, lane = tid & 31;
    const int m = lane & 15, hi = lane >> 4;   // A: row m, K-half hi
    const int kOff = hi << 1;

    const long token = (long)blockIdx.x * 128 + wave * 16 + m;
    const float* arow = x + token * D_MODEL + kOff;    // A(m, k0+kOff .. +1)
    const float* brow = sq + (m << 10);                // B col n == m (expert)
    const int    bXor = m << 2;

    v8f c = {0.f, 0.f, 0.f, 0.f, 0.f, 0.f, 0.f, 0.f};
    #pragma unroll 8
    for (int k0 = 0; k0 < D_MODEL; k0 += 4) {
        v2f a = *(const v2f*)(arow + k0);
        v2f b = *(const v2f*)(brow + ((k0 + kOff) ^ bXor));
        c = __builtin_amdgcn_wmma_f32_16x16x4_f32(
                false, a, false, b, (short)0, c, false, false);
    }

    __syncthreads();   // all waves done reading sq -> reuse as score scratch
    #pragma unroll
    for (int r = 0; r < 8; ++r) {
        int row = r + (hi << 3);                       // C: M = r + 8*hi
        sq[((wave << 4) + row) * 16 + m] = c[r];       // scores[tokLocal][expert]
    }
    __syncthreads();

    if (tid < 128) {
        float best = sq[tid * 16];
        int bi = 0;
        #pragma unroll
        for (int t = 1; t < N_TILES; ++t) {
            float v = sq[tid * 16 + t];
            if (v > best) { best = v; bi = t; }        // first-max tie-break
        }
        long tk = (long)blockIdx.x * 128 + tid;
        idx_out[tk] = (float)bi;
        ws[WS_EXPERTOF + tk] = bi;
        atomicAdd(&ws[WS_COUNTS + bi], 1);
    }
}

// ---------------------------------------------------------------------------
// Scan: pad each expert bucket to a multiple of 16, emit tile tables.
// ---------------------------------------------------------------------------
__global__ void scan_kernel(int* __restrict__ ws) {
    if (threadIdx.x != 0 || blockIdx.x != 0) return;
    int off = 0, T = 0;
    for (int t = 0; t < N_TILES; ++t) {
        int c = ws[WS_COUNTS + t];
        ws[WS_CURSORS + t] = off;
        int nt = (c + 15) >> 4;
        for (int i = 0; i < nt; ++i) {
            ws[WS_TILEEXP + T]  = t;
            ws[WS_TILEBASE + T] = off + (i << 4);
            ++T;
        }
        for (int j = c; j < (nt << 4); ++j) ws[WS_PERM + off + j] = -1;
        off += nt << 4;
    }
    ws[WS_NUMTILES] = T;
}

__global__ __launch_bounds__(256) void scatter_kernel(int* __restrict__ ws) {
    int b = blockIdx.x * 256 + threadIdx.x;
    int t = ws[WS_EXPERTOF + b];
    int pos = atomicAdd(&ws[WS_CURSORS + t], 1);
    ws[WS_PERM + pos] = b;
}

// ---------------------------------------------------------------------------
// Grouped GEMM: block = one 16-token tile x all 1024 columns of its expert.
// x rows staged into swizzled LDS via gfx1250 async-to-LDS copies (ASYNCcnt,
// no VGPR round trip); W streams from L2 (64 MB fits in the 192 MB L2);
// fp32 WMMA 16x16x4, K accumulated over 1024.
// ---------------------------------------------------------------------------
__global__ __launch_bounds__(256) void gemm_kernel(
    const float* __restrict__ x, const float* __restrict__ W,
    const float* __restrict__ bias, float* __restrict__ out,
    const int* __restrict__ ws)
{
    __shared__ float ldsA[16 * D_MODEL];   // 64 KB, XOR-swizzled rows
    const int tile = blockIdx.x;
    if (tile >= ws[WS_NUMTILES]) return;   // uniform exit: EXEC stays all-1s

    const int t    = ws[WS_TILEEXP + tile];
    const int base = ws[WS_TILEBASE + tile];
    const int* perm = ws + WS_PERM;
    const int tid = threadIdx.x;

    // Stage A: 16 rows x 1024 floats, one async b128 per (thread, row).
    // Output row m of the WMMA tile depends only on A row m, so padding rows
    // (perm == -1) may hold garbage: clamp the token to 0 and mask the store
    // in the epilogue. Keeps EXEC all-1s through staging (no divergence).
    const int k4 = tid << 2;                       // this thread's float4 col
    #pragma unroll
    for (int row = 0; row < 16; ++row) {
        int tok = perm[base + row];
        int tokc = tok < 0 ? 0 : tok;
        const float* gptr = x + (size_t)tokc * D_MODEL + k4;
        unsigned ldsOff = (unsigned)(size_t)(ldsA + (row << 10) + (k4 ^ (row << 2)));
        asm volatile("global_load_async_to_lds_b128 %0, %1, off"
                     :: "v"(ldsOff), "v"(gptr) : "memory");
    }
    asm volatile("s_wait_asynccnt 0x0" ::: "memory");
    __syncthreads();

    const int wave = tid >> 5, lane = tid & 31;
    const int m = lane & 15, hi = lane >> 4;
    const int kOff = hi << 1;
    const float* aBase = ldsA + (m << 10);
    const int    aXor  = m << 2;

    // Epilogue tokens: C-matrix row M = r + 8*hi (hoisted out of pass loop).
    int tokr[8];
    #pragma unroll
    for (int r = 0; r < 8; ++r) tokr[r] = perm[base + r + (hi << 3)];

    for (int pass = 0; pass < 8; ++pass) {
        const int col = (pass << 7) + (wave << 4) + m;   // B col n == m
        const float* wrow = W + ((size_t)t * N_CLASSES + col) * D_MODEL + kOff;

        v8f c = {0.f, 0.f, 0.f, 0.f, 0.f, 0.f, 0.f, 0.f};
        #pragma unroll 8
        for (int k0 = 0; k0 < D_MODEL; k0 += 4) {
            v2f a = *(const v2f*)(aBase + ((k0 + kOff) ^ aXor));  // ds_load_b64
            v2f b = *(const v2f*)(wrow + k0);                     // global b64
            c = __builtin_amdgcn_wmma_f32_16x16x4_f32(
                    false, a, false, b, (short)0, c, false, false);
        }

        const float bv = bias[t * N_CLASSES + col];
        #pragma unroll
        for (int r = 0; r < 8; ++r) {
            if (tokr[r] >= 0)
                out[(size_t)tokr[r] * N_CLASSES + col] = c[r] + bv;
        }
    }
}

extern "C" void kernel_launch(void* const* d_in, const int* in_sizes, int n_in,
                              void* d_out, int out_size, void* d_ws, size_t ws_size,
                              hipStream_t stream) {
    const float* x    = (const float*)d_in[0];
    const float* sigs = (const float*)d_in[1];
    const float* W    = (const float*)d_in[2];
    const float* b    = (const float*)d_in[3];

    float* logits  = (float*)d_out;                         // [BATCH, N_CLASSES]
    float* idx_out = logits + (size_t)BATCH * N_CLASSES;    // [BATCH] (as float)
    int*   ws      = (int*)d_ws;

    init_kernel   <<<1,           32,  0, stream>>>(ws);
    route_kernel  <<<BATCH / 128, 256, 0, stream>>>(x, sigs, idx_out, ws);
    scan_kernel   <<<1,           1,   0, stream>>>(ws);
    scatter_kernel<<<BATCH / 256, 256, 0, stream>>>(ws);
    gemm_kernel   <<<MAXT,        256, 0, stream>>>(x, W, b, logits, ws);
}